// CustomEmbedding_73770358276324
// MI455X (gfx1250) — compile-verified
//
#include <hip/hip_runtime.h>
#include <cstdint>
#include <cstddef>

// Embedding row-gather: out[s,:] = weights[ids[s],:]
//   S = 8192 tokens, D = 768 fp32  -> row = 3072 bytes
//
// Pure data movement (0 FLOPs): route it through the CDNA5 async
// global<->LDS copy path (ASYNCcnt) with B128 transactions.
//   - 192 threads/block (6 wave32), one block per token
//   - lane t moves bytes [16t, 16t+16) of the row: 192*16B = 3072B
//   - global_load_async_to_lds_b128   : HBM/L2 -> LDS (no VGPR data)
//   - s_wait_asynccnt 0               : per-wave completion of the load
//   - global_store_async_from_lds_b128: LDS -> HBM, non-temporal (th:TH_STORE_NT)
//
// NT store rationale: the 151.8 MB table fits in the 192 MB L2; the 25.2 MB
// output is write-once. Keeping output lines out of L2 preserves table
// residency so repeated token ids hit L2 instead of HBM.
//
// Each lane touches only its own 16B LDS slot, so no workgroup barrier
// is required; ASYNCcnt ordering is sufficient.

#define DIM        768
#define ROW_BYTES  (DIM * 4)         // 3072
#define THREADS    (ROW_BYTES / 16)  // 192 lanes * 16B = one row

typedef __attribute__((address_space(3))) char lds_char_t;

__global__ __launch_bounds__(THREADS)
void embed_gather_async_kernel(const int* __restrict__ ids,
                               const float* __restrict__ weights,
                               float* __restrict__ out)
{
    __shared__ char smem[ROW_BYTES];

    const int tid = threadIdx.x;           // 0..191
    const int s   = blockIdx.x;            // token index
    const int row = ids[s];                // wave-uniform -> scalar load

    const char* gsrc = (const char*)weights + (size_t)row * ROW_BYTES + (size_t)tid * 16;
    char*       gdst = (char*)out          + (size_t)s   * ROW_BYTES + (size_t)tid * 16;

    // 32-bit LDS byte address for this lane's slot (addrspace(3) pointer).
    lds_char_t* lp = (lds_char_t*)smem + tid * 16;

    // HBM/L2 -> LDS, async (counts on ASYNCcnt, no VGPR data movement).
    // Default (RT) load policy: keep table lines resident in L2.
    asm volatile("global_load_async_to_lds_b128 %0, %1, off"
                 :
                 : "v"(lp), "v"(gsrc)
                 : "memory");

    // Wait for this wave's async load to land in LDS.
    asm volatile("s_wait_asynccnt 0" ::: "memory");

    // LDS -> HBM, async, non-temporal store (write-once output must not
    // evict the L2-resident embedding table).
    asm volatile("global_store_async_from_lds_b128 %0, %1, off th:TH_STORE_NT"
                 :
                 : "v"(gdst), "v"(lp)
                 : "memory");

    // Drain the async store (s_endpgm also implies wait-idle; explicit for clarity).
    asm volatile("s_wait_asynccnt 0" ::: "memory");
}

extern "C" void kernel_launch(void* const* d_in, const int* in_sizes, int n_in,
                              void* d_out, int out_size, void* d_ws, size_t ws_size,
                              hipStream_t stream)
{
    const int*   ids     = (const int*)d_in[0];     // x: [1, SEQ] int32
    const float* weights = (const float*)d_in[1];   // [VOCAB, DIM] fp32
    float*       out     = (float*)d_out;           // [SEQ, DIM] fp32

    const int seq = in_sizes[0];                    // 8192

    embed_gather_async_kernel<<<seq, THREADS, 0, stream>>>(ids, weights, out);
}